// GaussianWithWeightAffinity_16484084483566
// MI455X (gfx1250) — compile-verified
//
#include <hip/hip_runtime.h>
#include <hip/hip_bf16.h>

typedef float v4f __attribute__((ext_vector_type(4)));
typedef int   v4i __attribute__((ext_vector_type(4)));

#define LOG2E 1.4426950408889634f

// ---------------------------------------------------------------------------
// Kernel 1: coef[b] = tanh(dot(Ws[b,:], A_w[0,:]) + A_b[0])   (b=2, K=64)
// Trivial cost; computed once into workspace so the big kernel doesn't redo
// a 64-FMA dot product per thread.
// ---------------------------------------------------------------------------
__global__ void gw_coef_kernel(const float* __restrict__ Ws,
                               const float* __restrict__ A_w,
                               const float* __restrict__ A_b,
                               float* __restrict__ coef,
                               int B, int K) {
    int b = threadIdx.x;
    if (b >= B) return;
    float acc = A_b[0];
    for (int k = 0; k < K; ++k) acc = fmaf(Ws[b * K + k], A_w[k], acc);
    coef[b] = tanhf(acc);
}

// ---------------------------------------------------------------------------
// Kernel 2: Me[b,i,j] = exp(-(Xs[b,i]*coef[b] - Ys[b,j])^2), NaN dist -> 0.
// Store-bandwidth bound (512 MiB out). Each 256-thread block computes a
// R x COLS tile: Y tile staged once via async global->LDS (CDNA5 ASYNCcnt
// path), reused for R rows; output written with 128-bit non-temporal stores.
// ---------------------------------------------------------------------------
template <int N, int R, int COLS>
__global__ __launch_bounds__(256) void gw_affinity_kernel(
    const float* __restrict__ Xs,
    const float* __restrict__ Ys,
    const float* __restrict__ coef,
    float* __restrict__ out) {
    static_assert(COLS == 256 * 4, "one float4 of Y per thread");
    __shared__ float ytile[COLS];

    constexpr int CHUNKS = N / COLS;           // column chunks per row (8)
    const int tid   = threadIdx.x;
    const int blk   = blockIdx.x;
    const int chunk = blk % CHUNKS;
    const int rowg  = blk / CHUNKS;
    const int row0  = rowg * R;                // global row in [0, B*N)
    const int b     = row0 / N;                // R divides N -> uniform b
    const int j0    = chunk * COLS + tid * 4;  // 16B-aligned column start

    const float* ysrc = Ys + (size_t)b * N + j0;

#if __has_builtin(__builtin_amdgcn_global_load_async_to_lds_b128)
    // gfx1250 async DMA: global -> LDS, tracked by ASYNCcnt.
    // Builtin signature (from hipcc diagnostic): (v4i addrspace(1)*,
    // v4i addrspace(3)*, imm offset, imm cpol).
    typedef __attribute__((address_space(1))) v4i gv4i_t;
    typedef __attribute__((address_space(3))) v4i lv4i_t;
    __builtin_amdgcn_global_load_async_to_lds_b128(
        (gv4i_t*)ysrc, (lv4i_t*)&ytile[tid * 4], /*offset=*/0, /*cpol=*/0);
    asm volatile("s_wait_asynccnt 0" ::: "memory");
    v4f y = *(const v4f*)&ytile[tid * 4];
#else
    // Fallback: plain 128-bit global load (hits L2, Y is only 64 KB total).
    v4f y = *(const v4f*)ysrc;
#endif

    const float cf = coef[b];
    float xr[R];
#pragma unroll
    for (int r = 0; r < R; ++r) xr[r] = Xs[row0 + r] * cf;

#pragma unroll
    for (int r = 0; r < R; ++r) {
        v4f o;
#pragma unroll
        for (int c = 0; c < 4; ++c) {
            float diff = xr[r] - y[c];
            float t    = diff * diff;                        // dist (SIGMA=1, AE=1)
            float e    = __builtin_amdgcn_exp2f(-LOG2E * t); // v_exp_f32
            o[c]       = (t != t) ? 0.0f : e;                // NaN -> inf -> exp = 0
        }
        // Streaming 512 MiB: non-temporal 128-bit store, don't pollute L2.
        __builtin_nontemporal_store(o, (v4f*)(out + (size_t)(row0 + r) * N + j0));
    }
}

extern "C" void kernel_launch(void* const* d_in, const int* in_sizes, int n_in,
                              void* d_out, int out_size, void* d_ws, size_t ws_size,
                              hipStream_t stream) {
    (void)in_sizes; (void)n_in; (void)out_size; (void)ws_size;
    const float* Xs  = (const float*)d_in[0];  // (2, 8192, 1)
    const float* Ys  = (const float*)d_in[1];  // (2, 8192, 1)
    const float* Ws  = (const float*)d_in[2];  // (2, 64)
    const float* A_w = (const float*)d_in[3];  // (1, 64)
    const float* A_b = (const float*)d_in[4];  // (1,)
    float* out  = (float*)d_out;               // (2, 8192, 8192)
    float* coef = (float*)d_ws;                // 2 floats scratch

    constexpr int B = 2, N = 8192, K = 64;
    constexpr int R = 4, COLS = 1024;

    gw_coef_kernel<<<1, 32, 0, stream>>>(Ws, A_w, A_b, coef, B, K);

    const int blocks = (B * N / R) * (N / COLS);   // 32768 blocks x 8 waves
    gw_affinity_kernel<N, R, COLS><<<blocks, 256, 0, stream>>>(Xs, Ys, coef, out);
}